// NystromAttention_85014582657248
// MI455X (gfx1250) — compile-verified
//
#include <hip/hip_runtime.h>

typedef __attribute__((ext_vector_type(2))) float v2f;
typedef __attribute__((ext_vector_type(8))) float v8f;

constexpr int Bc = 4, Nc = 4096, DIMc = 512, Hc = 8, DHc = 64, Mc = 256, Lc = 16, KWc = 33;

// ---------------------------------------------------------------------------
// f32 WMMA GEMM mainloop: block tile 128x64, BK=16, 256 threads = 8 wave32s.
// Waves arranged 4 (rows) x 2 (cols); each wave owns a 32x32 slab as a 2x2
// grid of 16x16 V_WMMA_F32_16X16X4_F32 accumulators chained over K.
// Double-buffered LDS with register staging: one barrier per K-stage, global
// loads for stage s+1 issued before the compute of stage s.
// Both As (row-major) and Bs (n-major) keep k,k+1 adjacent -> b64 LDS loads
// straight into even-aligned VGPR pairs for WMMA operands (no mov shuffles).
// ---------------------------------------------------------------------------
__device__ __forceinline__ void gemm_tile_f32(
    const float* __restrict__ A, int lda,
    const float* __restrict__ B, int ldb,
    int Kd, int rowBase, int colBase, bool transB,
    v8f acc[2][2])
{
    __shared__ float As[2][128][20];   // row stride 80B (16B aligned)
    __shared__ float Bs[2][64][20];    // n-major: Bs[n][k], row stride 80B

    const int tid  = threadIdx.x;
    const int lane = tid & 31;
    const int wave = tid >> 5;
    const int wy   = wave >> 1;   // 0..3
    const int wx   = wave & 1;    // 0..1

    v8f zero = {};
    acc[0][0] = zero; acc[0][1] = zero; acc[1][0] = zero; acc[1][1] = zero;

    const int arow = wy * 32 + (lane & 15);
    const int bcol = wx * 32 + (lane & 15);
    const int kq   = (lane >> 4) << 1;      // 0 or 2

    const int ar  = tid >> 1;               // 0..127   (A: 2 float4 per thread)
    const int acg = (tid & 1) << 3;         // 0 or 8
    const int bk  = tid >> 4;               // 0..15    (B notrans: float4 along N)
    const int bn  = (tid & 15) << 2;        // 0..60
    const int tn  = tid & 63;               // B trans: float4 along K
    const int tkg = (tid >> 6) << 2;        // 0,4,8,12

    const float* apBase = A + (size_t)(rowBase + ar) * lda + acg;

    float4 a0r, a1r, br;
    // prologue: stage 0 global -> regs -> LDS buffer 0
    a0r = *(const float4*)apBase;
    a1r = *(const float4*)(apBase + 4);
    if (!transB) br = *(const float4*)(B + (size_t)bk * ldb + colBase + bn);
    else         br = *(const float4*)(B + (size_t)(colBase + tn) * ldb + tkg);

    *(float4*)&As[0][ar][acg]     = a0r;
    *(float4*)&As[0][ar][acg + 4] = a1r;
    if (!transB) {
        Bs[0][bn + 0][bk] = br.x; Bs[0][bn + 1][bk] = br.y;
        Bs[0][bn + 2][bk] = br.z; Bs[0][bn + 3][bk] = br.w;
    } else {
        *(float4*)&Bs[0][tn][tkg] = br;
    }
    __syncthreads();

    const int nStages = Kd >> 4;
    for (int s = 0; s < nStages; ++s) {
        const int buf  = s & 1;
        const bool more = (s + 1) < nStages;
        const int k0n  = (s + 1) << 4;

        if (more) {  // prefetch next stage into registers
            a0r = *(const float4*)(apBase + k0n);
            a1r = *(const float4*)(apBase + k0n + 4);
            if (!transB) br = *(const float4*)(B + (size_t)(k0n + bk) * ldb + colBase + bn);
            else         br = *(const float4*)(B + (size_t)(colBase + tn) * ldb + k0n + tkg);
        }

        #pragma unroll
        for (int ks = 0; ks < 16; ks += 4) {
            v2f a0, a1, b0, b1;
            a0[0] = As[buf][arow][ks + kq];        a0[1] = As[buf][arow][ks + kq + 1];
            a1[0] = As[buf][arow + 16][ks + kq];   a1[1] = As[buf][arow + 16][ks + kq + 1];
            b0[0] = Bs[buf][bcol][ks + kq];        b0[1] = Bs[buf][bcol][ks + kq + 1];
            b1[0] = Bs[buf][bcol + 16][ks + kq];   b1[1] = Bs[buf][bcol + 16][ks + kq + 1];
            acc[0][0] = __builtin_amdgcn_wmma_f32_16x16x4_f32(false, a0, false, b0,
                                                              (short)0, acc[0][0], false, false);
            acc[0][1] = __builtin_amdgcn_wmma_f32_16x16x4_f32(false, a0, false, b1,
                                                              (short)0, acc[0][1], false, false);
            acc[1][0] = __builtin_amdgcn_wmma_f32_16x16x4_f32(false, a1, false, b0,
                                                              (short)0, acc[1][0], false, false);
            acc[1][1] = __builtin_amdgcn_wmma_f32_16x16x4_f32(false, a1, false, b1,
                                                              (short)0, acc[1][1], false, false);
        }

        if (more) {  // drain registers into the other LDS buffer
            const int nb = buf ^ 1;
            *(float4*)&As[nb][ar][acg]     = a0r;
            *(float4*)&As[nb][ar][acg + 4] = a1r;
            if (!transB) {
                Bs[nb][bn + 0][bk] = br.x; Bs[nb][bn + 1][bk] = br.y;
                Bs[nb][bn + 2][bk] = br.z; Bs[nb][bn + 3][bk] = br.w;
            } else {
                *(float4*)&Bs[nb][tn][tkg] = br;
            }
        }
        __syncthreads();
    }
}

__device__ __forceinline__ void epi_coords(int rowBase, int colBase, int& row0, int& col0)
{
    int tid  = threadIdx.x;
    int lane = tid & 31;
    int wave = tid >> 5;
    row0 = rowBase + (wave >> 1) * 32 + ((lane >> 4) << 3);  // + mi*16 + r
    col0 = colBase + (wave & 1) * 32 + (lane & 15);          // + ni*16
}

// ---------------------------------------------------------------------------
// 1) QKV projection with fused reshape to [b,h,n,d] and q *= dh^-0.5
// ---------------------------------------------------------------------------
__global__ __launch_bounds__(256) void k_qkv(
    const float* __restrict__ x, const float* __restrict__ W,
    float* __restrict__ q, float* __restrict__ k, float* __restrict__ v)
{
    int rowBase = blockIdx.x * 128;
    int colBase = blockIdx.y * 64;
    v8f acc[2][2];
    gemm_tile_f32(x, DIMc, W, 3 * Hc * DHc, DIMc, rowBase, colBase, false, acc);
    int row0, col0; epi_coords(rowBase, colBase, row0, col0);
    #pragma unroll
    for (int ni = 0; ni < 2; ++ni) {
        int col = col0 + ni * 16;
        int which = col / (Hc * DHc);
        int hc = col % (Hc * DHc);
        int h = hc / DHc, d = hc % DHc;
        float* dst = (which == 0) ? q : (which == 1) ? k : v;
        float scale = (which == 0) ? 0.125f : 1.0f;   // DH^-0.5 = 64^-0.5
        #pragma unroll
        for (int mi = 0; mi < 2; ++mi) {
            #pragma unroll
            for (int r = 0; r < 8; ++r) {
                int row = row0 + mi * 16 + r;
                int b = row / Nc, n = row % Nc;
                dst[(((size_t)(b * Hc + h)) * Nc + n) * DHc + d] = acc[mi][ni][r] * scale;
            }
        }
    }
}

// 2) landmark means over windows of Lc=16
__global__ void k_landmark(const float* __restrict__ q, const float* __restrict__ k,
                           float* __restrict__ qlm, float* __restrict__ klm)
{
    int t = blockIdx.x * 256 + threadIdx.x;
    if (t >= Bc * Hc * Mc * DHc) return;
    int d  = t % DHc;
    int m  = (t / DHc) % Mc;
    int bh = t / (DHc * Mc);
    const float* qb = q + ((size_t)bh * Nc + m * Lc) * DHc + d;
    const float* kb = k + ((size_t)bh * Nc + m * Lc) * DHc + d;
    float sq = 0.0f, sk = 0.0f;
    #pragma unroll
    for (int j = 0; j < Lc; ++j) { sq += qb[(size_t)j * DHc]; sk += kb[(size_t)j * DHc]; }
    qlm[t] = sq * (1.0f / Lc);
    klm[t] = sk * (1.0f / Lc);
}

// 3) generic batched GEMM: D = A @ B(^T)
__global__ __launch_bounds__(256) void k_gemm_bn(
    const float* __restrict__ A, size_t sA, int lda,
    const float* __restrict__ B, size_t sB, int ldb,
    float* __restrict__ D, size_t sD, int ldd,
    int Kd, int transB)
{
    int bh = blockIdx.z;
    const float* Ab = A + (size_t)bh * sA;
    const float* Bb = B + (size_t)bh * sB;
    float*       Db = D + (size_t)bh * sD;
    int rowBase = blockIdx.x * 128, colBase = blockIdx.y * 64;
    v8f acc[2][2];
    gemm_tile_f32(Ab, lda, Bb, ldb, Kd, rowBase, colBase, transB != 0, acc);
    int row0, col0; epi_coords(rowBase, colBase, row0, col0);
    #pragma unroll
    for (int mi = 0; mi < 2; ++mi)
        #pragma unroll
        for (int ni = 0; ni < 2; ++ni) {
            int col = col0 + ni * 16;
            #pragma unroll
            for (int r = 0; r < 8; ++r) {
                int row = row0 + mi * 16 + r;
                Db[(size_t)row * ldd + col] = acc[mi][ni][r];
            }
        }
}

// 4) softmax over rows of length rowLen (in place)
__global__ void k_softmax(float* __restrict__ data, int rowLen)
{
    __shared__ float red[256];
    float* row = data + (size_t)blockIdx.x * rowLen;
    int tid = threadIdx.x;
    float m = -3.402823466e38f;
    for (int i = tid; i < rowLen; i += 256) m = fmaxf(m, row[i]);
    red[tid] = m; __syncthreads();
    for (int s = 128; s > 0; s >>= 1) {
        if (tid < s) red[tid] = fmaxf(red[tid], red[tid + s]);
        __syncthreads();
    }
    m = red[0]; __syncthreads();
    float sum = 0.0f;
    for (int i = tid; i < rowLen; i += 256) { float e = __expf(row[i] - m); row[i] = e; sum += e; }
    red[tid] = sum; __syncthreads();
    for (int s = 128; s > 0; s >>= 1) {
        if (tid < s) red[tid] += red[tid + s];
        __syncthreads();
    }
    float inv = 1.0f / red[0];
    for (int i = tid; i < rowLen; i += 256) row[i] *= inv;
}

// 5) pinv init: global row/col abs-sums of attn2, then z0 = attn2^T / (maxc*maxr)
__global__ void k_colrow_sum(const float* __restrict__ a2,
                             float* __restrict__ cs, float* __restrict__ rs)
{
    int t = blockIdx.x * 256 + threadIdx.x;
    if (t >= Bc * Hc * Mc) return;
    int bh = t / Mc, j = t % Mc;
    const float* base = a2 + (size_t)bh * Mc * Mc;
    float c = 0.0f, r = 0.0f;
    for (int i = 0; i < Mc; ++i) {
        c += fabsf(base[(size_t)i * Mc + j]);
        r += fabsf(base[(size_t)j * Mc + i]);
    }
    cs[t] = c; rs[t] = r;
}

__global__ void k_scal(const float* __restrict__ cs, const float* __restrict__ rs,
                       float* __restrict__ scal)
{
    __shared__ float red[256];
    float mc = 0.0f, mr = 0.0f;
    for (int i = threadIdx.x; i < Bc * Hc * Mc; i += 256) {
        mc = fmaxf(mc, cs[i]); mr = fmaxf(mr, rs[i]);
    }
    red[threadIdx.x] = mc; __syncthreads();
    for (int s = 128; s > 0; s >>= 1) {
        if (threadIdx.x < s) red[threadIdx.x] = fmaxf(red[threadIdx.x], red[threadIdx.x + s]);
        __syncthreads();
    }
    mc = red[0]; __syncthreads();
    red[threadIdx.x] = mr; __syncthreads();
    for (int s = 128; s > 0; s >>= 1) {
        if (threadIdx.x < s) red[threadIdx.x] = fmaxf(red[threadIdx.x], red[threadIdx.x + s]);
        __syncthreads();
    }
    mr = red[0];
    if (threadIdx.x == 0) scal[0] = 1.0f / (mc * mr);
}

__global__ void k_zinit(const float* __restrict__ a2, const float* __restrict__ scal,
                        float* __restrict__ z)
{
    int t = blockIdx.x * 256 + threadIdx.x;
    if (t >= Bc * Hc * Mc * Mc) return;
    int j = t % Mc; int i = (t / Mc) % Mc; int bh = t / (Mc * Mc);
    z[t] = a2[((size_t)bh * Mc + j) * Mc + i] * scal[0];
}

// 6) pinv GEMM step: D = alpha*(A@B) + beta*I + gamma*Cin   (256x256, batched)
__global__ __launch_bounds__(256) void k_pinv_gemm(
    const float* __restrict__ A, const float* __restrict__ B,
    const float* __restrict__ Cin, float* __restrict__ D,
    float alpha, float beta, float gamma)
{
    int bh = blockIdx.z;
    size_t s = (size_t)Mc * Mc;
    const float* Ab = A + bh * s;
    const float* Bb = B + bh * s;
    const float* Cb = Cin ? Cin + bh * s : nullptr;
    float*       Db = D + bh * s;
    int rowBase = blockIdx.x * 128, colBase = blockIdx.y * 64;
    v8f acc[2][2];
    gemm_tile_f32(Ab, Mc, Bb, Mc, Mc, rowBase, colBase, false, acc);
    int row0, col0; epi_coords(rowBase, colBase, row0, col0);
    #pragma unroll
    for (int mi = 0; mi < 2; ++mi)
        #pragma unroll
        for (int ni = 0; ni < 2; ++ni) {
            int col = col0 + ni * 16;
            #pragma unroll
            for (int r = 0; r < 8; ++r) {
                int row = row0 + mi * 16 + r;
                float val = alpha * acc[mi][ni][r];
                if (row == col) val += beta;
                if (Cb) val += gamma * Cb[(size_t)row * Mc + col];
                Db[(size_t)row * Mc + col] = val;
            }
        }
}

// 7) (attn1@z) @ (attn3@v) with scatter into [b, n, h*DH + d] layout
__global__ __launch_bounds__(256) void k_out_scatter(
    const float* __restrict__ A,   // [bh, N, M]
    const float* __restrict__ Bm,  // [bh, M, DH]
    float* __restrict__ outb)      // [B*N, H*DH]
{
    int bh = blockIdx.z; int b = bh / Hc, h = bh % Hc;
    const float* Ab = A  + (size_t)bh * Nc * Mc;
    const float* Bb = Bm + (size_t)bh * Mc * DHc;
    int rowBase = blockIdx.x * 128, colBase = blockIdx.y * 64;
    v8f acc[2][2];
    gemm_tile_f32(Ab, Mc, Bb, DHc, Mc, rowBase, colBase, false, acc);
    int row0, col0; epi_coords(rowBase, colBase, row0, col0);
    #pragma unroll
    for (int mi = 0; mi < 2; ++mi)
        #pragma unroll
        for (int ni = 0; ni < 2; ++ni) {
            int col = col0 + ni * 16;
            #pragma unroll
            for (int r = 0; r < 8; ++r) {
                int row = row0 + mi * 16 + r;
                outb[((size_t)(b * Nc + row)) * (Hc * DHc) + h * DHc + col] = acc[mi][ni][r];
            }
        }
}

// 8) depthwise residual conv along seq, accumulated into outb
__global__ void k_conv_add(const float* __restrict__ v, const float* __restrict__ rk,
                           float* __restrict__ outb)
{
    int t = blockIdx.x * 256 + threadIdx.x;
    if (t >= Bc * Hc * Nc * DHc) return;
    int d = t % DHc;
    int n = (t / DHc) % Nc;
    int h = (t / (DHc * Nc)) % Hc;
    int b = t / (DHc * Nc * Hc);
    const float* vb = v + ((size_t)(b * Hc + h) * Nc) * DHc + d;
    float acc = 0.0f;
    #pragma unroll
    for (int j = 0; j < KWc; ++j) {
        int nn = n + j - KWc / 2;
        if (nn >= 0 && nn < Nc) acc += vb[(size_t)nn * DHc] * rk[h * KWc + j];
    }
    outb[((size_t)(b * Nc + n)) * (Hc * DHc) + h * DHc + d] += acc;
}

// 9) final projection: out = outb @ Wout + bout
__global__ __launch_bounds__(256) void k_final(
    const float* __restrict__ A, const float* __restrict__ W,
    const float* __restrict__ bias, float* __restrict__ out)
{
    int rowBase = blockIdx.x * 128, colBase = blockIdx.y * 64;
    v8f acc[2][2];
    gemm_tile_f32(A, DIMc, W, DIMc, DIMc, rowBase, colBase, false, acc);
    int row0, col0; epi_coords(rowBase, colBase, row0, col0);
    #pragma unroll
    for (int mi = 0; mi < 2; ++mi)
        #pragma unroll
        for (int ni = 0; ni < 2; ++ni) {
            int col = col0 + ni * 16;
            float bv = bias[col];
            #pragma unroll
            for (int r = 0; r < 8; ++r) {
                int row = row0 + mi * 16 + r;
                out[(size_t)row * DIMc + col] = acc[mi][ni][r] + bv;
            }
        }
}

// ---------------------------------------------------------------------------
extern "C" void kernel_launch(void* const* d_in, const int* in_sizes, int n_in,
                              void* d_out, int out_size, void* d_ws, size_t ws_size,
                              hipStream_t stream)
{
    const float* x    = (const float*)d_in[0];
    const float* Wqkv = (const float*)d_in[1];
    const float* Wout = (const float*)d_in[2];
    const float* bout = (const float*)d_in[3];
    const float* rk   = (const float*)d_in[4];
    float* out = (float*)d_out;
    float* ws  = (float*)d_ws;

    size_t o = 0;
    float* Q    = ws + o; o += (size_t)Bc * Hc * Nc * DHc;
    float* Kt   = ws + o; o += (size_t)Bc * Hc * Nc * DHc;
    float* V    = ws + o; o += (size_t)Bc * Hc * Nc * DHc;
    float* QLM  = ws + o; o += (size_t)Bc * Hc * Mc * DHc;
    float* KLM  = ws + o; o += (size_t)Bc * Hc * Mc * DHc;
    float* A1   = ws + o; o += (size_t)Bc * Hc * Nc * Mc;   // attn1 [bh,N,M]
    float* A2   = ws + o; o += (size_t)Bc * Hc * Mc * Mc;   // attn2 [bh,M,M]
    float* A3   = ws + o; o += (size_t)Bc * Hc * Mc * Nc;   // attn3 [bh,M,N]; reused for attn1@z
    float* Z0   = ws + o; o += (size_t)Bc * Hc * Mc * Mc;
    float* Z1   = ws + o; o += (size_t)Bc * Hc * Mc * Mc;
    float* XZ   = ws + o; o += (size_t)Bc * Hc * Mc * Mc;
    float* T2   = ws + o; o += (size_t)Bc * Hc * Mc * Mc;
    float* T3   = ws + o; o += (size_t)Bc * Hc * Mc * Mc;
    float* A3V  = ws + o; o += (size_t)Bc * Hc * Mc * DHc;
    float* OUTB = ws + o; o += (size_t)Bc * Nc * Hc * DHc;
    float* CS   = ws + o; o += (size_t)Bc * Hc * Mc;
    float* RS   = ws + o; o += (size_t)Bc * Hc * Mc;
    float* SCAL = ws + o; o += 1;

    const int BH = Bc * Hc;
    dim3 blk(256);

    // QKV projection + reshape + q-scale
    k_qkv<<<dim3((Bc * Nc) / 128, (3 * Hc * DHc) / 64), blk, 0, stream>>>(x, Wqkv, Q, Kt, V);

    // landmarks
    k_landmark<<<dim3((Bc * Hc * Mc * DHc + 255) / 256), blk, 0, stream>>>(Q, Kt, QLM, KLM);

    // sim1 = q @ k_lm^T   [bh, N, M]
    k_gemm_bn<<<dim3(Nc / 128, Mc / 64, BH), blk, 0, stream>>>(
        Q, (size_t)Nc * DHc, DHc, KLM, (size_t)Mc * DHc, DHc,
        A1, (size_t)Nc * Mc, Mc, DHc, 1);
    // sim2 = q_lm @ k_lm^T [bh, M, M]
    k_gemm_bn<<<dim3(Mc / 128, Mc / 64, BH), blk, 0, stream>>>(
        QLM, (size_t)Mc * DHc, DHc, KLM, (size_t)Mc * DHc, DHc,
        A2, (size_t)Mc * Mc, Mc, DHc, 1);
    // sim3 = q_lm @ k^T    [bh, M, N]
    k_gemm_bn<<<dim3(Mc / 128, Nc / 64, BH), blk, 0, stream>>>(
        QLM, (size_t)Mc * DHc, DHc, Kt, (size_t)Nc * DHc, DHc,
        A3, (size_t)Mc * Nc, Nc, DHc, 1);

    // softmaxes
    k_softmax<<<dim3(BH * Nc), blk, 0, stream>>>(A1, Mc);
    k_softmax<<<dim3(BH * Mc), blk, 0, stream>>>(A2, Mc);
    k_softmax<<<dim3(BH * Mc), blk, 0, stream>>>(A3, Nc);

    // pinv init: z0 = attn2^T / (max(rowsums)*max(colsums))
    k_colrow_sum<<<dim3((BH * Mc + 255) / 256), blk, 0, stream>>>(A2, CS, RS);
    k_scal<<<dim3(1), blk, 0, stream>>>(CS, RS, SCAL);
    k_zinit<<<dim3((BH * Mc * Mc + 255) / 256), blk, 0, stream>>>(A2, SCAL, Z0);

    // 6 Newton-Schulz iterations: z' = 0.25 z (13I - xz(15I - xz(7I - xz)))
    float* zin = Z0; float* zout = Z1;
    dim3 g44(Mc / 128, Mc / 64, BH);
    for (int it = 0; it < 6; ++it) {
        k_pinv_gemm<<<g44, blk, 0, stream>>>(A2, zin, nullptr, XZ, 1.0f, 0.0f, 0.0f);
        k_pinv_gemm<<<g44, blk, 0, stream>>>(XZ, XZ, XZ, T2, 1.0f, 15.0f, -7.0f); // xz^2-7xz+15I
        k_pinv_gemm<<<g44, blk, 0, stream>>>(XZ, T2, nullptr, T3, -1.0f, 13.0f, 0.0f);
        k_pinv_gemm<<<g44, blk, 0, stream>>>(zin, T3, nullptr, zout, 0.25f, 0.0f, 0.0f);
        float* t = zin; zin = zout; zout = t;
    }

    // attn3 @ v -> [bh, M, DH]
    k_gemm_bn<<<dim3(Mc / 128, DHc / 64, BH), blk, 0, stream>>>(
        A3, (size_t)Mc * Nc, Nc, V, (size_t)Nc * DHc, DHc,
        A3V, (size_t)Mc * DHc, DHc, Nc, 0);
    // attn1 @ z -> reuse A3 buffer as [bh, N, M]
    k_gemm_bn<<<dim3(Nc / 128, Mc / 64, BH), blk, 0, stream>>>(
        A1, (size_t)Nc * Mc, Mc, zin, (size_t)Mc * Mc, Mc,
        A3, (size_t)Nc * Mc, Mc, Mc, 0);
    // (attn1 z) @ (attn3 v) -> scattered to [B*N, H*DH]
    k_out_scatter<<<dim3(Nc / 128, 1, BH), blk, 0, stream>>>(A3, A3V, OUTB);
    // depthwise residual conv added in
    k_conv_add<<<dim3((Bc * Hc * Nc * DHc + 255) / 256), blk, 0, stream>>>(V, rk, OUTB);
    // final projection + bias
    k_final<<<dim3((Bc * Nc) / 128, DIMc / 64), blk, 0, stream>>>(OUTB, Wout, bout, out);
}